// GCN_6176162972388
// MI455X (gfx1250) — compile-verified
//
#include <hip/hip_runtime.h>
#include <hip/hip_bf16.h>

typedef float v2f __attribute__((ext_vector_type(2)));
typedef float v4f __attribute__((ext_vector_type(4)));
typedef float v8f __attribute__((ext_vector_type(8)));

// ---------------------------------------------------------------------------
// Degree / normalization kernels
// ---------------------------------------------------------------------------
__global__ void gcn_zero_f32(float* __restrict__ p, int n) {
    int i = blockIdx.x * blockDim.x + threadIdx.x;
    if (i < n) p[i] = 0.0f;
}

__global__ void gcn_deg_count(const long long* __restrict__ dst,
                              float* __restrict__ deg, int E) {
    int i = blockIdx.x * blockDim.x + threadIdx.x;
    if (i < E) atomicAdd(&deg[(int)dst[i]], 1.0f);
}

__global__ void gcn_dinv(float* __restrict__ deg, int n) {
    int i = blockIdx.x * blockDim.x + threadIdx.x;
    if (i < n) deg[i] = rsqrtf(deg[i] + 1.0f);   // +1 for the self loop
}

// ---------------------------------------------------------------------------
// fp32 WMMA GEMM with K-pair-interleaved LDS-staged B + fused epilogue.
//   H[M,N] = A[M,K] * B[K,N]
//   O[M,N] = H * dinv[row]^2 (+ bias[col] if bias != nullptr)
//
// LDS layout: pair-row p holds K-rows {2p, 2p+1} interleaved element-wise:
//   bs[p*S2 + 2c + 0] = B[2p  ][c]
//   bs[p*S2 + 2c + 1] = B[2p+1][c]
// so a lane's B fragment {B[k][c], B[k+1][c]} is ONE aligned ds_load_b64
// straight into the even VGPR pair the WMMA consumes (no register shuffles).
// S2 = 2N+32 dwords == 32 (mod 64): kh=0 lanes use banks c..c+31 and kh=2
// lanes banks c+32..c+63 -> conflict-free.
//
// Block = 256 threads (8 waves); each wave computes a 16x64 C strip:
// one A fragment (global b64) feeds 4 WMMAs (16x16x4 f32) per K-step.
// ---------------------------------------------------------------------------
template <int K, int N>
__global__ void __launch_bounds__(256)
gcn_gemm_fused(const float* __restrict__ A, const float* __restrict__ B,
               float* __restrict__ H, float* __restrict__ O,
               const float* __restrict__ dinv, const float* __restrict__ bias,
               int M) {
    constexpr int S2   = 2 * N + 32;         // dwords per K-pair row in LDS
    constexpr int NBLK = N / 64;             // 64-wide N subtiles per row
    __shared__ float bs[(K / 2) * S2];

    // ---- cooperative stage of B into LDS (interleaved K pairs, b128 stores)
    constexpr int TASKS = (K / 2) * (N / 2);
    for (int i = threadIdx.x; i < TASKS; i += 256) {
        int p  = i / (N / 2);
        int c  = (i - p * (N / 2)) << 1;
        v2f r0 = *(const v2f*)(B + (size_t)(2 * p)     * N + c);
        v2f r1 = *(const v2f*)(B + (size_t)(2 * p + 1) * N + c);
        v4f w;
        w.x = r0.x; w.y = r1.x; w.z = r0.y; w.w = r1.y;
        *(v4f*)&bs[p * S2 + (c << 1)] = w;
    }
    __syncthreads();

    const int lane = threadIdx.x & 31;
    const int wave = blockIdx.x * 8 + (threadIdx.x >> 5);
    const int tm   = wave / NBLK;
    const int tn   = wave - tm * NBLK;
    if (tm * 16 >= M) return;

    const int l15 = lane & 15;
    const int kh  = (lane >> 4) << 1;        // 0 or 2

    const float* aptr = A + (size_t)(tm * 16 + l15) * K + kh;
    int boff = (kh >> 1) * S2 + ((tn * 64 + l15) << 1);

    v8f acc0 = {}, acc1 = {}, acc2 = {}, acc3 = {};
#pragma unroll 4
    for (int k = 0; k < K; k += 4) {
        v2f a  = *(const v2f*)aptr;                  // K = kh+k, kh+k+1
        v2f b0 = *(const v2f*)&bs[boff +  0];        // cols +0,+16,+32,+48
        v2f b1 = *(const v2f*)&bs[boff + 32];
        v2f b2 = *(const v2f*)&bs[boff + 64];
        v2f b3 = *(const v2f*)&bs[boff + 96];
        acc0 = __builtin_amdgcn_wmma_f32_16x16x4_f32(false, a, false, b0,
                                                     (short)0, acc0, false, false);
        acc1 = __builtin_amdgcn_wmma_f32_16x16x4_f32(false, a, false, b1,
                                                     (short)0, acc1, false, false);
        acc2 = __builtin_amdgcn_wmma_f32_16x16x4_f32(false, a, false, b2,
                                                     (short)0, acc2, false, false);
        acc3 = __builtin_amdgcn_wmma_f32_16x16x4_f32(false, a, false, b3,
                                                     (short)0, acc3, false, false);
        aptr += 4;
        boff += 2 * S2;                      // advance two K-pair rows
    }

    // ---- fused epilogue: write H and self-loop-initialized O ----
    const int col = tn * 64 + l15;
    const int rw0 = tm * 16 + ((lane >> 4) << 3);
#pragma unroll
    for (int r = 0; r < 8; ++r) {
        const int row = rw0 + r;
        const float dv = dinv[row];
        const float w  = dv * dv;
        float h0 = acc0[r], h1 = acc1[r], h2 = acc2[r], h3 = acc3[r];
        float* hp = H + (size_t)row * N + col;
        hp[0] = h0; hp[16] = h1; hp[32] = h2; hp[48] = h3;
        float* op = O + (size_t)row * N + col;
        if (bias) {
            op[0]  = h0 * w + bias[col +  0];
            op[16] = h1 * w + bias[col + 16];
            op[32] = h2 * w + bias[col + 32];
            op[48] = h3 * w + bias[col + 48];
        } else {
            op[0] = h0 * w; op[16] = h1 * w; op[32] = h2 * w; op[48] = h3 * w;
        }
    }
}

// ---------------------------------------------------------------------------
// Edge scatter: out[d,f] += H[s,f] * dinv[s]*dinv[d], 4 features per thread.
// F is a compile-time power of two -> index math is shifts/masks.
// ---------------------------------------------------------------------------
template <int F>
__global__ void gcn_scatter_edges(const float* __restrict__ H,
                                  const float* __restrict__ dinv,
                                  const long long* __restrict__ src,
                                  const long long* __restrict__ dst,
                                  float* __restrict__ out, int E) {
    constexpr int FQ = F / 4;
    int idx = blockIdx.x * blockDim.x + threadIdx.x;
    if (idx >= E * FQ) return;
    int e = idx / FQ;                        // power-of-two -> shift
    int f = (idx & (FQ - 1)) << 2;
    int s = (int)src[e];
    int d = (int)dst[e];
    float w = dinv[s] * dinv[d];
    v4f h = *(const v4f*)(H + (size_t)s * F + f);
    float* o = out + (size_t)d * F + f;
    atomicAdd(o + 0, h.x * w);
    atomicAdd(o + 1, h.y * w);
    atomicAdd(o + 2, h.z * w);
    atomicAdd(o + 3, h.w * w);
}

// ---------------------------------------------------------------------------
// In-place bias + ReLU (F compile-time power of two)
// ---------------------------------------------------------------------------
template <int F>
__global__ void gcn_bias_relu(float* __restrict__ A,
                              const float* __restrict__ bias, int total) {
    int idx = blockIdx.x * blockDim.x + threadIdx.x;
    if (idx >= total) return;
    float v = A[idx] + bias[idx & (F - 1)];
    A[idx] = v > 0.0f ? v : 0.0f;
}

// ---------------------------------------------------------------------------
// Host-side launcher
// ---------------------------------------------------------------------------
extern "C" void kernel_launch(void* const* d_in, const int* in_sizes, int n_in,
                              void* d_out, int out_size, void* d_ws, size_t ws_size,
                              hipStream_t stream) {
    const float*     x   = (const float*)d_in[0];
    const long long* ei  = (const long long*)d_in[1];
    const float*     W1  = (const float*)d_in[2];
    const float*     b1  = (const float*)d_in[3];
    const float*     W2  = (const float*)d_in[4];
    const float*     b2  = (const float*)d_in[5];

    const int M = in_sizes[0] / 128;      // 50000 nodes
    const int E = in_sizes[1] / 2;        // 800000 edges

    const long long* src = ei;
    const long long* dst = ei + E;

    // Workspace layout (~103 MB):
    //   [0, 256KB)         : dinv (M floats)
    //   [256KB, +M*256*4)  : H  (GEMM outputs; reused for layer 2)
    //   [.. , +M*256*4)    : A1 (aggregation buffer / layer-1 activations)
    char*  ws   = (char*)d_ws;
    float* dinv = (float*)ws;
    float* H    = (float*)(ws + (size_t)256 * 1024);
    float* A1   = (float*)(ws + (size_t)256 * 1024 + (size_t)M * 256 * 4);

    const int T = 256;
    auto blocks = [](long long n, int t) { return (unsigned)((n + t - 1) / t); };

    // Degree -> dinv
    gcn_zero_f32<<<blocks(M, T), T, 0, stream>>>(dinv, M);
    gcn_deg_count<<<blocks(E, T), T, 0, stream>>>(dst, dinv, E);
    gcn_dinv<<<blocks(M, T), T, 0, stream>>>(dinv, M);

    // ---- Layer 1: H = x@W1 ; A1 = H*dinv^2 ; edge scatter ; relu(A1+b1)
    {
        long long waves = (long long)((M + 15) / 16) * (256 / 64);
        gcn_gemm_fused<128, 256><<<blocks(waves * 32, T), T, 0, stream>>>(
            x, W1, H, A1, dinv, nullptr, M);
        gcn_scatter_edges<256><<<blocks((long long)E * 64, T), T, 0, stream>>>(
            H, dinv, src, dst, A1, E);
        gcn_bias_relu<256><<<blocks((long long)M * 256, T), T, 0, stream>>>(
            A1, b1, M * 256);
    }

    // ---- Layer 2: H = A1@W2 ; out = H*dinv^2 + b2 ; edge scatter
    {
        float* out = (float*)d_out;
        long long waves = (long long)((M + 15) / 16) * (128 / 64);
        gcn_gemm_fused<256, 128><<<blocks(waves * 32, T), T, 0, stream>>>(
            A1, W2, H, out, dinv, b2, M);
        gcn_scatter_edges<128><<<blocks((long long)E * 32, T), T, 0, stream>>>(
            H, dinv, src, dst, out, E);
    }
}